// GCN_46969762349063
// MI455X (gfx1250) — compile-verified
//
#include <hip/hip_runtime.h>
#include <hip/hip_bf16.h>

typedef __attribute__((ext_vector_type(2))) float v2f;
typedef __attribute__((ext_vector_type(8))) float v8f;

#define HID 16
#define NCLS 10

// ---------------------------------------------------------------------------
// Kernel 1: init deg = 1.0 (self-loop weight), s = 0
// ---------------------------------------------------------------------------
__global__ void gcn_init_kernel(float* __restrict__ deg, float* __restrict__ s, int n) {
    int i = blockIdx.x * blockDim.x + threadIdx.x;
    if (i < n) {
        deg[i] = 1.0f;
        s[i]   = 0.0f;
    }
}

// ---------------------------------------------------------------------------
// Kernel 2: deg[col] += w  (segment_sum of edge weights at target nodes)
// ---------------------------------------------------------------------------
__global__ void gcn_deg_kernel(const long long* __restrict__ ei,
                               const float* __restrict__ w,
                               float* __restrict__ deg, int e) {
    int i = blockIdx.x * blockDim.x + threadIdx.x;
    if (i < e) {
        int col = (int)ei[(size_t)e + i];
        atomicAdd(&deg[col], w[i]);
    }
}

// ---------------------------------------------------------------------------
// Kernel 3: dis = deg^{-1/2}   (deg >= 1 always, so no zero-guard needed)
// ---------------------------------------------------------------------------
__global__ void gcn_dis_kernel(const float* __restrict__ deg,
                               float* __restrict__ dis, int n) {
    int i = blockIdx.x * blockDim.x + threadIdx.x;
    if (i < n) dis[i] = rsqrtf(deg[i]);
}

// ---------------------------------------------------------------------------
// Kernel 4: layer-1 aggregation, factored to a SCALAR per edge:
//   s[col] += dis[row]*w*dis[col] * x[row]
// (valid because h1 = x @ W1 is rank-1: aggregation commutes with * W1[j])
// ---------------------------------------------------------------------------
__global__ void gcn_scatter1_kernel(const long long* __restrict__ ei,
                                    const float* __restrict__ w,
                                    const float* __restrict__ x,
                                    const float* __restrict__ dis,
                                    float* __restrict__ s, int e) {
    int i = blockIdx.x * blockDim.x + threadIdx.x;
    if (i < e) {
        int row = (int)ei[i];
        int col = (int)ei[(size_t)e + i];
        float norm = dis[row] * w[i] * dis[col];
        atomicAdd(&s[col], norm * x[row]);
    }
}

// ---------------------------------------------------------------------------
// Kernel 5: h[n][j] = relu( (s[n] + dis[n]^2 * x[n]) * W1[j] + b1[j] )
// one thread per (node, feature)
// ---------------------------------------------------------------------------
__global__ void gcn_h_kernel(const float* __restrict__ s,
                             const float* __restrict__ dis,
                             const float* __restrict__ x,
                             const float* __restrict__ W1,
                             const float* __restrict__ b1,
                             float* __restrict__ h, int n) {
    int i = blockIdx.x * blockDim.x + threadIdx.x;
    if (i < n * HID) {
        int node = i >> 4;
        int j    = i & 15;
        float d  = dis[node];
        float t  = s[node] + d * d * x[node];
        float v  = t * W1[j] + b1[j];
        h[i] = v > 0.0f ? v : 0.0f;
    }
}

// ---------------------------------------------------------------------------
// Kernel 6: g = h @ W2pad  via V_WMMA_F32_16X16X4_F32 (exact fp32).
// One wave per 16-node tile; K=16 consumed as 4 chained k=4 WMMAs.
// A 16x4 layout (ISA 7.12.2): lane M = lane&15; lanes 0-15 hold K={0,1},
// lanes 16-31 hold K={2,3} in VGPRs {0,1}.  B 4x16 mirrored (row=K striped
// across lanes).  C/D: VGPR r holds row M = r + 8*(lane>=16), col N = lane&15.
//
// W2 zero-padding: unconditional loads at a CLAMPED in-bounds address, then
// v_cndmask zeroing (no EXEC predication in the compute section).
// Stores: wave-uniform full-tile fast path with one base pointer and 8
// constant-offset global_store_b32 (rows differ by a fixed 64B stride).
// ---------------------------------------------------------------------------
__global__ __launch_bounds__(256) void gcn_gemm_wmma_kernel(
        const float* __restrict__ h,
        const float* __restrict__ W2,
        float* __restrict__ g, int n) {
    int wave = (blockIdx.x * blockDim.x + threadIdx.x) >> 5;
    int lane = threadIdx.x & 31;
    int nTiles = (n + 15) >> 4;
    if (wave >= nTiles) return;            // wave-uniform exit: EXEC stays full

    int m0     = wave << 4;
    int laneLo = lane & 15;
    int khalf  = (lane >> 4) << 1;         // 0 for lanes 0-15, 2 for lanes 16-31
    int rowA   = m0 + laneLo;
    if (rowA > n - 1) rowA = n - 1;        // clamp (keeps EXEC all-1s)

    bool valid = laneLo < NCLS;
    int  colB  = valid ? laneLo : (NCLS - 1);   // clamped, always in-bounds

    // Fetch the entire padded B tile (16x16, 2 elems/lane/k-step) up front.
    v2f b[4];
    #pragma unroll
    for (int kk = 0; kk < 4; ++kk) {
        int k0 = (kk << 2) + khalf;
        float b0 = W2[k0 * NCLS + colB];
        float b1 = W2[(k0 + 1) * NCLS + colB];
        b[kk].x = valid ? b0 : 0.0f;
        b[kk].y = valid ? b1 : 0.0f;
    }

    v8f c = {};
    #pragma unroll
    for (int kk = 0; kk < 4; ++kk) {
        int k0 = (kk << 2) + khalf;
        v2f a;
        a.x = h[(size_t)rowA * HID + k0];
        a.y = h[(size_t)rowA * HID + k0 + 1];
        c = __builtin_amdgcn_wmma_f32_16x16x4_f32(
                /*neg_a=*/false, a, /*neg_b=*/false, b[kk],
                /*c_mod=*/(short)0, c, /*reuse_a=*/false, /*reuse_b=*/false);
    }

    int    mOff = (lane >> 4) << 3;        // 0 or 8
    float* gp   = g + (size_t)(m0 + mOff) * HID + laneLo;
    if (m0 + 16 <= n) {
        // full tile: wave-uniform branch, 8 constant-offset stores
        #pragma unroll
        for (int i = 0; i < 8; ++i) gp[i * HID] = c[i];
    } else {
        // tail tile (never taken when N % 16 == 0): per-lane guards
        #pragma unroll
        for (int i = 0; i < 8; ++i) {
            if (m0 + mOff + i < n) gp[i * HID] = c[i];
        }
    }
}

// ---------------------------------------------------------------------------
// Kernel 7: out[n][j] = dis[n]^2 * g[n][j] + b2[j]   (self-loop + bias init)
// ---------------------------------------------------------------------------
__global__ void gcn_out_init_kernel(const float* __restrict__ g,
                                    const float* __restrict__ dis,
                                    const float* __restrict__ b2,
                                    float* __restrict__ out, int n) {
    int i = blockIdx.x * blockDim.x + threadIdx.x;
    if (i < n * NCLS) {
        int node = i / NCLS;
        int j    = i - node * NCLS;
        float d  = dis[node];
        out[i] = d * d * g[(size_t)node * HID + j] + b2[j];
    }
}

// ---------------------------------------------------------------------------
// Kernel 8: layer-2 aggregation: out[col][0..9] += norm * g[row][0..9]
// g rows are 64B-aligned -> two float4 + one float2 loads (all hit L2: 6.4MB)
// ---------------------------------------------------------------------------
__global__ void gcn_scatter2_kernel(const long long* __restrict__ ei,
                                    const float* __restrict__ w,
                                    const float* __restrict__ g,
                                    const float* __restrict__ dis,
                                    float* __restrict__ out, int e) {
    int i = blockIdx.x * blockDim.x + threadIdx.x;
    if (i < e) {
        int row = (int)ei[i];
        int col = (int)ei[(size_t)e + i];
        float norm = dis[row] * w[i] * dis[col];
        const float4* gr = (const float4*)(g + (size_t)row * HID);
        float4 f0 = gr[0];
        float4 f1 = gr[1];
        float2 f2 = *(const float2*)(g + (size_t)row * HID + 8);
        float* o = out + (size_t)col * NCLS;
        atomicAdd(o + 0, norm * f0.x);
        atomicAdd(o + 1, norm * f0.y);
        atomicAdd(o + 2, norm * f0.z);
        atomicAdd(o + 3, norm * f0.w);
        atomicAdd(o + 4, norm * f1.x);
        atomicAdd(o + 5, norm * f1.y);
        atomicAdd(o + 6, norm * f1.z);
        atomicAdd(o + 7, norm * f1.w);
        atomicAdd(o + 8, norm * f2.x);
        atomicAdd(o + 9, norm * f2.y);
    }
}

// ---------------------------------------------------------------------------
extern "C" void kernel_launch(void* const* d_in, const int* in_sizes, int n_in,
                              void* d_out, int out_size, void* d_ws, size_t ws_size,
                              hipStream_t stream) {
    const float*     x   = (const float*)d_in[0];       // [N,1]
    const long long* ei  = (const long long*)d_in[1];   // [2,E] int64
    const float*     ew  = (const float*)d_in[2];       // [E]
    const float*     W1  = (const float*)d_in[3];       // [1,16]
    const float*     b1  = (const float*)d_in[4];       // [16]
    const float*     W2  = (const float*)d_in[5];       // [16,10]
    const float*     b2  = (const float*)d_in[6];       // [10]
    float*           out = (float*)d_out;               // [N,10]

    const int N = in_sizes[0];
    const int E = in_sizes[2];

    // workspace carve-out: (3 + 32)*N floats = ~14 MB for N=100k
    float* deg = (float*)d_ws;            // [N]
    float* dis = deg + N;                 // [N]
    float* s   = dis + N;                 // [N]
    float* h   = s   + N;                 // [N,16]
    float* g   = h   + (size_t)N * HID;   // [N,16] (cols 10..15 are padding)

    const int TB = 256;
    dim3 blkN((N + TB - 1) / TB);
    dim3 blkE((E + TB - 1) / TB);
    dim3 blkNH((N * HID + TB - 1) / TB);
    dim3 blkNC((N * NCLS + TB - 1) / TB);
    int nTiles = (N + 15) >> 4;
    dim3 blkT((nTiles * 32 + TB - 1) / TB);

    gcn_init_kernel    <<<blkN,  TB, 0, stream>>>(deg, s, N);
    gcn_deg_kernel     <<<blkE,  TB, 0, stream>>>(ei, ew, deg, E);
    gcn_dis_kernel     <<<blkN,  TB, 0, stream>>>(deg, dis, N);
    gcn_scatter1_kernel<<<blkE,  TB, 0, stream>>>(ei, ew, x, dis, s, E);
    gcn_h_kernel       <<<blkNH, TB, 0, stream>>>(s, dis, x, W1, b1, h, N);
    gcn_gemm_wmma_kernel<<<blkT, TB, 0, stream>>>(h, W2, g, N);
    gcn_out_init_kernel<<<blkNC, TB, 0, stream>>>(g, dis, b2, out, N);
    gcn_scatter2_kernel<<<blkE,  TB, 0, stream>>>(ei, ew, g, dis, out, E);
}